// TransformerLayer_with_bond_10273561772407
// MI455X (gfx1250) — compile-verified
//
#include <hip/hip_runtime.h>
#include <hip/hip_bf16.h>
#include <math.h>

#define NNODES 10000
#define NEDGES 100000

typedef __attribute__((ext_vector_type(2))) float v2f;
typedef __attribute__((ext_vector_type(8))) float v8f;

__device__ __forceinline__ v8f wmma4(v2f a, v2f b, v8f c) {
  // D = A(16x4 f32) x B(4x16 f32) + C(16x16 f32)
  return __builtin_amdgcn_wmma_f32_16x16x4_f32(false, a, false, b, (short)0, c,
                                               false, false);
}
__device__ __forceinline__ v8f vzero8() {
  v8f r;
  #pragma unroll
  for (int i = 0; i < 8; ++i) r[i] = 0.f;
  return r;
}
__device__ __forceinline__ float frcp(float xx) {  // single v_rcp_f32
  return __builtin_amdgcn_rcpf(xx);
}
__device__ __forceinline__ float sus(float xx) {  // soft_unit_step
  return (xx > 0.f) ? __expf(-frcp(xx)) : 0.f;
}
__device__ __forceinline__ float silu(float xx) {
  return xx * frcp(1.f + __expf(-xx));
}

// ---------------- workspace layout (float offsets) ----------------
#define WS_Z 0              // [10000]  softmax denominators
#define WS_Q 10016          // [160000] q = xs@Wq/sqrt32
#define WS_EXPV 170016      // [100000]
#define WS_VS 270016        // [E*32]
#define WS_VV 3470016       // [E*24]
#define WS_W1K 5870016      // [2048] packed
#define WS_W1V 5872064      // [2048]
#define WS_WDOT 5874112     // [256]
#define WS_W2K 5874368      // [81920]
#define WS_W2V 5956288      // [212992]

// Pack row-major W[K][NCOL] (K = nkb*4) into per-lane WMMA B-operand order:
// float2 at tid=(c*nkb+kb)*32+l  holds W[kb*4+2*(l>>4)+r][c*16+(l&15)]*scale
__global__ void pack_w_kernel(const float* __restrict__ W, float* __restrict__ o,
                              int NCOL, int nchunk, int nkb, float scale) {
  int tid = blockIdx.x * blockDim.x + threadIdx.x;
  if (tid >= nchunk * nkb * 32) return;
  int l = tid & 31;
  int kb = (tid >> 5) % nkb;
  int c = tid / (nkb * 32);
  int col = c * 16 + (l & 15);
  int row = kb * 4 + 2 * (l >> 4);
  o[tid * 2 + 0] = W[row * NCOL + col] * scale;
  o[tid * 2 + 1] = W[(row + 1) * NCOL + col] * scale;
}

// q = xs @ Wq / sqrt(32); also zero the z accumulator
__global__ void node_q_kernel(const float* __restrict__ x,
                              const float* __restrict__ Wq,
                              float* __restrict__ q, float* __restrict__ z) {
  int tid = blockIdx.x * blockDim.x + threadIdx.x;
  if (tid >= NNODES * 16) return;
  int n = tid >> 4, c = tid & 15;
  const float* xr = x + n * 56;
  float acc = 0.f;
  #pragma unroll
  for (int u = 0; u < 32; ++u) acc += xr[u] * Wq[u * 16 + c];
  q[tid] = acc * 0.17677669529663687f;
  if (c == 0) z[n] = 0.f;
}

// out[n,w] = si_s  (scalar self-interaction), w<32
__global__ void node_sis_kernel(const float* __restrict__ x,
                                const float* __restrict__ na,
                                const float* __restrict__ Ws,
                                float* __restrict__ out) {
  int tid = blockIdx.x * blockDim.x + threadIdx.x;
  if (tid >= NNODES * 32) return;
  int n = tid >> 5, w = tid & 31;
  const float* xr = x + n * 56;
  const float* nr = na + n * 16;
  float acc = 0.f;
  for (int u = 0; u < 32; ++u) {
    const float* Wr = Ws + u * 512 + w;
    float s = 0.f;
    #pragma unroll
    for (int v = 0; v < 16; ++v) s += nr[v] * Wr[v * 32];
    acc += xr[u] * s;
  }
  out[n * 56 + w] = acc * 0.04419417382415922f;  // 1/sqrt(512)
}

// out[n,32+w*3+i] = si_v  (vector self-interaction)
__global__ void node_siv_kernel(const float* __restrict__ x,
                                const float* __restrict__ na,
                                const float* __restrict__ Wv,
                                float* __restrict__ out) {
  int tid = blockIdx.x * blockDim.x + threadIdx.x;
  if (tid >= NNODES * 24) return;
  int n = tid / 24, t = tid % 24;
  int w = t / 3, i = t % 3;
  const float* xr = x + n * 56 + 32;
  const float* nr = na + n * 16;
  float acc = 0.f;
  #pragma unroll
  for (int u = 0; u < 8; ++u) {
    const float* Wr = Wv + u * 128 + w;
    float s = 0.f;
    #pragma unroll
    for (int v = 0; v < 16; ++v) s += nr[v] * Wr[v * 8];
    acc += xr[u * 3 + i] * s;
  }
  out[n * 56 + 32 + t] = acc * 0.08838834764831845f;  // 1/sqrt(128)
}

#define HSTR 132  // padded LDS stride for H (kills bank conflicts)

__global__ __launch_bounds__(32) void edge_kernel(
    const float* __restrict__ x, const float* __restrict__ pos,
    const float* __restrict__ eattr, const int* __restrict__ eidx,
    const float* __restrict__ qv, const float* __restrict__ W1kp,
    const float* __restrict__ W1vp, const float* __restrict__ W2kp,
    const float* __restrict__ W2vp, const float* __restrict__ Wdp,
    float* __restrict__ z, float* __restrict__ expv_g,
    float* __restrict__ VSg, float* __restrict__ VVg) {
  __shared__ float ED[16 * 18];    // edge_data, padded stride
  __shared__ float S[40 * 16];     // xs/sqrt32 (u<32), (xv.sh1)/sqrt24 (u>=32)
  __shared__ float Vb[24 * 16];    // xv/sqrt8        [u*3+i][e]
  __shared__ float MEb[24 * 16];   // (M2 xv)/sqrt8   [u*3+i][e]
  __shared__ float Hk[16 * HSTR];  // hidden activations, k-path
  __shared__ float Hv[16 * HSTR];
  __shared__ float P[32 * 8];      // attention partials
  __shared__ float T1[16 * 8], T2[3][16 * 8], T3[3][16 * 8];

  const int L = threadIdx.x;
  const int l16 = L & 15, hl = L >> 4, lb3 = (L >> 3) & 1;
  const int tile = blockIdx.x;
  const int ecol = tile * 16 + l16;
  const int dstc = eidx[NEDGES + ecol];

  float sh1v[3] = {0.f, 0.f, 0.f};
  float cutw = 0.f;

  if (L < 16) {  // one lane per edge: geometry + per-edge scales
    const int srcn = eidx[ecol];
    float e0 = pos[srcn * 3 + 0] - pos[dstc * 3 + 0];
    float e1 = pos[srcn * 3 + 1] - pos[dstc * 3 + 1];
    float e2 = pos[srcn * 3 + 2] - pos[dstc * 3 + 2];
    float r = sqrtf(e0 * e0 + e1 * e1 + e2 * e2 + 1e-12f);
    const float step = 6.f / 9.f;       // linspace(0,6,10)[1:-1] spacing
    const float istep = 1.5f;           // exact reciprocal of step
    const float embC = 1.14136f * 7.3890560989306495f * 2.8284271247461903f;
    #pragma unroll
    for (int k = 0; k < 8; ++k) {
      float d = (r - (k + 1) * step) * istep;
      ED[l16 * 18 + k] = embC * sus(d + 1.f) * sus(1.f - d);
    }
    #pragma unroll
    for (int k = 0; k < 8; ++k) ED[l16 * 18 + 8 + k] = eattr[ecol * 8 + k];
    cutw = sus(10.f * (1.f - r * (1.f / 6.f)));
    float ir = frcp(r);
    float nx = e0 * ir, ny = e1 * ir, nz = e2 * ir;
    const float s3 = 1.7320508075688772f;
    sh1v[0] = s3 * nx; sh1v[1] = s3 * ny; sh1v[2] = s3 * nz;
    float nn[3] = {nx, ny, nz};
    float M2m[3][3];
    #pragma unroll
    for (int i = 0; i < 3; ++i)
      #pragma unroll
      for (int j = 0; j < 3; ++j)
        M2m[i][j] = nn[i] * nn[j] - ((i == j) ? (1.f / 3.f) : 0.f);
    const float* xr = x + srcn * 56;
    const float is32 = 0.17677669529663687f, is24 = 0.2041241452319315f,
                is8 = 0.35355339059327373f;
    #pragma unroll
    for (int u = 0; u < 32; ++u) S[u * 16 + l16] = xr[u] * is32;
    #pragma unroll
    for (int u = 0; u < 8; ++u) {
      float a0 = xr[32 + u * 3], a1 = xr[33 + u * 3], a2 = xr[34 + u * 3];
      S[(32 + u) * 16 + l16] =
          (a0 * sh1v[0] + a1 * sh1v[1] + a2 * sh1v[2]) * is24;
      Vb[(u * 3 + 0) * 16 + l16] = a0 * is8;
      Vb[(u * 3 + 1) * 16 + l16] = a1 * is8;
      Vb[(u * 3 + 2) * 16 + l16] = a2 * is8;
      #pragma unroll
      for (int i = 0; i < 3; ++i)
        MEb[(u * 3 + i) * 16 + l16] =
            (M2m[i][0] * a0 + M2m[i][1] * a1 + M2m[i][2] * a2) * is8;
    }
  }
  __syncthreads();

  // ---- GEMM1: h = silu(edata @ W1/sqrt16), both paths ----
  v2f Ae[4];
  #pragma unroll
  for (int kb = 0; kb < 4; ++kb)
    Ae[kb] = *(const v2f*)&ED[l16 * 18 + 4 * kb + 2 * hl];
  #pragma unroll 1
  for (int path = 0; path < 2; ++path) {
    const float* Wp = path ? W1vp : W1kp;
    float* H = path ? Hv : Hk;
    #pragma unroll 1
    for (int c = 0; c < 8; ++c) {
      v8f acc = vzero8();
      #pragma unroll
      for (int kb = 0; kb < 4; ++kb) {
        v2f b = *(const v2f*)&Wp[((c * 4 + kb) * 32 + L) * 2];
        acc = wmma4(Ae[kb], b, acc);
      }
      #pragma unroll
      for (int r = 0; r < 8; ++r)
        H[(r + 8 * hl) * HSTR + c * 16 + l16] = silu(acc[r]);
    }
  }
  __syncthreads();

  // ---- GEMM2 k-path: 40 column chunks, fused contraction with S ----
  v2f Ah[32];
  #pragma unroll
  for (int kb = 0; kb < 32; ++kb)
    Ah[kb] = *(const v2f*)&Hk[l16 * HSTR + 4 * kb + 2 * hl];
  v8f k_acc = vzero8();
  #pragma unroll 1
  for (int c = 0; c < 40; ++c) {
    v8f d = vzero8();
    const float* bp = W2kp + (c * 1024 + L) * 2;
    #pragma unroll
    for (int kb = 0; kb < 32; ++kb)
      d = wmma4(Ah[kb], *(const v2f*)(bp + kb * 64), d);
    const float* sp = &S[c * 16 + hl * 8];
    #pragma unroll
    for (int r = 0; r < 8; ++r) k_acc[r] += sp[r] * d[r];
  }

  // ---- GEMM2 v-path: 104 chunks -> vs, t1, t2[i], t3[i] ----
  #pragma unroll
  for (int kb = 0; kb < 32; ++kb)
    Ah[kb] = *(const v2f*)&Hv[l16 * HSTR + 4 * kb + 2 * hl];
  v8f vs_lo = vzero8(), vs_hi = vzero8(), t1 = vzero8();
  v8f t2a[3], t3a[3];
  #pragma unroll
  for (int i = 0; i < 3; ++i) { t2a[i] = vzero8(); t3a[i] = vzero8(); }

  #pragma unroll 1
  for (int c = 0; c < 80; ++c) {  // wvA (0..63) + wvB (64..79) -> vs
    v8f d = vzero8();
    const float* bp = W2vp + (c * 1024 + L) * 2;
    #pragma unroll
    for (int kb = 0; kb < 32; ++kb)
      d = wmma4(Ah[kb], *(const v2f*)(bp + kb * 64), d);
    int su = (c < 64) ? (c >> 1) : (32 + ((c - 64) >> 1));
    const float* sp = &S[su * 16 + hl * 8];
    if (c & 1) {
      #pragma unroll
      for (int r = 0; r < 8; ++r) vs_hi[r] += sp[r] * d[r];
    } else {
      #pragma unroll
      for (int r = 0; r < 8; ++r) vs_lo[r] += sp[r] * d[r];
    }
  }
  #pragma unroll 1
  for (int c = 80; c < 96; ++c) {  // wvC -> t1 (scaled by xs)
    v8f d = vzero8();
    const float* bp = W2vp + (c * 1024 + L) * 2;
    #pragma unroll
    for (int kb = 0; kb < 32; ++kb)
      d = wmma4(Ah[kb], *(const v2f*)(bp + kb * 64), d);
    const float* sp = &S[(2 * (c - 80) + lb3) * 16 + hl * 8];
    #pragma unroll
    for (int r = 0; r < 8; ++r) t1[r] += sp[r] * d[r];
  }
  #pragma unroll 1
  for (int c = 96; c < 100; ++c) {  // wvD -> t2[i] (scaled by xv[:,i])
    v8f d = vzero8();
    const float* bp = W2vp + (c * 1024 + L) * 2;
    #pragma unroll
    for (int kb = 0; kb < 32; ++kb)
      d = wmma4(Ah[kb], *(const v2f*)(bp + kb * 64), d);
    #pragma unroll
    for (int i = 0; i < 3; ++i) {
      const float* sp = &Vb[((2 * (c - 96) + lb3) * 3 + i) * 16 + hl * 8];
      #pragma unroll
      for (int r = 0; r < 8; ++r) t2a[i][r] += sp[r] * d[r];
    }
  }
  #pragma unroll 1
  for (int c = 100; c < 104; ++c) {  // wvE -> t3[i] (scaled by M2 xv)
    v8f d = vzero8();
    const float* bp = W2vp + (c * 1024 + L) * 2;
    #pragma unroll
    for (int kb = 0; kb < 32; ++kb)
      d = wmma4(Ah[kb], *(const v2f*)(bp + kb * 64), d);
    #pragma unroll
    for (int i = 0; i < 3; ++i) {
      const float* sp = &MEb[((2 * (c - 100) + lb3) * 3 + i) * 16 + hl * 8];
      #pragma unroll
      for (int r = 0; r < 8; ++r) t3a[i][r] += sp[r] * d[r];
    }
  }

  // ---- attention: qw = q[dst] @ Wdot/16 via WMMA; temp = <qw, k> ----
  v8f qw = vzero8();
  {
    const float* qp = qv + dstc * 16;
    #pragma unroll
    for (int kb = 0; kb < 4; ++kb) {
      v2f a = *(const v2f*)&qp[4 * kb + 2 * hl];
      v2f b = *(const v2f*)&Wdp[(kb * 32 + L) * 2];
      qw = wmma4(a, b, qw);
    }
  }
  #pragma unroll
  for (int r = 0; r < 8; ++r) P[L * 8 + r] = qw[r] * k_acc[r];
  __syncthreads();
  if (L < 16) {
    int base = (L < 8) ? 0 : 16, rr = L & 7;
    float ts = 0.f;
    #pragma unroll
    for (int l = 0; l < 16; ++l) ts += P[(base + l) * 8 + rr];
    float ev = __expf(cutw * ts);
    expv_g[ecol] = ev;
    atomicAdd(&z[dstc], ev);
  }

  // ---- store vs ----
  #pragma unroll
  for (int r = 0; r < 8; ++r) {
    int e = tile * 16 + r + 8 * hl;
    VSg[e * 32 + l16] = vs_lo[r];
    VSg[e * 32 + 16 + l16] = vs_hi[r];
  }

  // ---- fold even/odd-u halves (lanes w and w+8) and assemble vv ----
  #pragma unroll
  for (int r = 0; r < 8; ++r) {
    t1[r] += __shfl_xor(t1[r], 8, 32);
    #pragma unroll
    for (int i = 0; i < 3; ++i) {
      t2a[i][r] += __shfl_xor(t2a[i][r], 8, 32);
      t3a[i][r] += __shfl_xor(t3a[i][r], 8, 32);
    }
  }
  if ((L & 8) == 0) {
    int w = L & 7;
    #pragma unroll
    for (int r = 0; r < 8; ++r) {
      int idx = (r + 8 * hl) * 8 + w;
      T1[idx] = t1[r];
      #pragma unroll
      for (int i = 0; i < 3; ++i) { T2[i][idx] = t2a[i][r]; T3[i][idx] = t3a[i][r]; }
    }
  }
  __syncthreads();
  if (L < 16) {
    int e = tile * 16 + L;
    #pragma unroll
    for (int w = 0; w < 8; ++w) {
      float a1 = T1[L * 8 + w];
      #pragma unroll
      for (int i = 0; i < 3; ++i)
        VVg[e * 24 + w * 3 + i] = sh1v[i] * a1 + T2[i][L * 8 + w] + T3[i][L * 8 + w];
    }
  }
}

// a = sqrt(expv/z + 1e-14); out[dst] += a * {vs, vv}
__global__ void scatter_kernel(const int* __restrict__ eidx,
                               const float* __restrict__ z,
                               const float* __restrict__ expv,
                               const float* __restrict__ VS,
                               const float* __restrict__ VV,
                               float* __restrict__ out) {
  int tid = blockIdx.x * blockDim.x + threadIdx.x;
  if (tid >= NEDGES * 64) return;
  int e = tid >> 6, c = tid & 63;
  if (c >= 56) return;
  int dst = eidx[NEDGES + e];
  float zz = z[dst];
  zz = (zz == 0.f) ? 1.f : zz;
  float a = sqrtf(expv[e] * frcp(zz) + 1e-14f);
  float v = (c < 32) ? VS[e * 32 + c] : VV[e * 24 + (c - 32)];
  atomicAdd(&out[dst * 56 + c], a * v);
}

extern "C" void kernel_launch(void* const* d_in, const int* in_sizes, int n_in,
                              void* d_out, int out_size, void* d_ws,
                              size_t ws_size, hipStream_t stream) {
  const float* x = (const float*)d_in[0];
  const float* pos = (const float*)d_in[1];
  const float* na = (const float*)d_in[2];
  const float* ea = (const float*)d_in[3];
  const float* Wq = (const float*)d_in[4];
  const float* Wsis = (const float*)d_in[5];
  const float* Wsiv = (const float*)d_in[6];
  const float* W1k = (const float*)d_in[7];
  const float* W2k = (const float*)d_in[8];
  const float* W1v = (const float*)d_in[9];
  const float* W2v = (const float*)d_in[10];
  const float* Wdot = (const float*)d_in[11];
  const int* eidx = (const int*)d_in[12];
  float* out = (float*)d_out;
  float* ws = (float*)d_ws;

  const float is128 = 0.08838834764831845f;
  pack_w_kernel<<<(8 * 4 * 32 + 255) / 256, 256, 0, stream>>>(
      W1k, ws + WS_W1K, 128, 8, 4, 0.25f);
  pack_w_kernel<<<(8 * 4 * 32 + 255) / 256, 256, 0, stream>>>(
      W1v, ws + WS_W1V, 128, 8, 4, 0.25f);
  pack_w_kernel<<<(1 * 4 * 32 + 255) / 256, 256, 0, stream>>>(
      Wdot, ws + WS_WDOT, 16, 1, 4, 0.0625f);
  pack_w_kernel<<<(40 * 32 * 32 + 255) / 256, 256, 0, stream>>>(
      W2k, ws + WS_W2K, 640, 40, 32, is128);
  pack_w_kernel<<<(104 * 32 * 32 + 255) / 256, 256, 0, stream>>>(
      W2v, ws + WS_W2V, 1664, 104, 32, is128);

  node_q_kernel<<<(NNODES * 16 + 255) / 256, 256, 0, stream>>>(
      x, Wq, ws + WS_Q, ws + WS_Z);
  node_sis_kernel<<<(NNODES * 32 + 255) / 256, 256, 0, stream>>>(x, na, Wsis, out);
  node_siv_kernel<<<(NNODES * 24 + 255) / 256, 256, 0, stream>>>(x, na, Wsiv, out);

  edge_kernel<<<NEDGES / 16, 32, 0, stream>>>(
      x, pos, ea, eidx, ws + WS_Q, ws + WS_W1K, ws + WS_W1V, ws + WS_W2K,
      ws + WS_W2V, ws + WS_WDOT, ws + WS_Z, ws + WS_EXPV, ws + WS_VS,
      ws + WS_VV);

  scatter_kernel<<<(NEDGES * 64 + 255) / 256, 256, 0, stream>>>(
      eidx, ws + WS_Z, ws + WS_EXPV, ws + WS_VS, ws + WS_VV, out);
}